// DepthwiseConvOverTimeLayer_15831249453291
// MI455X (gfx1250) — compile-verified
//
#include <hip/hip_runtime.h>

// B=16, T=20, H=7, W=7, C=1024, M=4, KD=KP=3
#define BB 16
#define TT 20
#define HH 7
#define WWD 7
#define CC 1024
#define MM 4

typedef float v2f __attribute__((ext_vector_type(2)));
typedef float v8f __attribute__((ext_vector_type(8)));

// ---------------------------------------------------------------------------
// Kernel 1: depthwise 3x3 conv (multiplier 4) fused with temporal max over T.
// ymax layout: [b][h][w][k], k = c*4 + m  (flat dw-channel order), f32.
// One thread per (b,h,w,c): 180 coalesced x loads, 720 FMAs, float4 store.
// ---------------------------------------------------------------------------
__global__ __launch_bounds__(256) void dw_tmax_kernel(
    const float* __restrict__ x,      // (B,T,H,W,C)
    const float* __restrict__ dw_w,   // (3,3,C,M)
    const float* __restrict__ dw_b,   // (C*M)
    float* __restrict__ ymax)         // (B,H,W,C*M)
{
    const int c  = blockIdx.x * blockDim.x + threadIdx.x;  // 0..1023
    const int hw = blockIdx.y;
    const int h  = hw / WWD, w = hw % WWD;
    const int b  = blockIdx.z;

    // 9 taps x 4 multipliers, contiguous float4 per tap
    float wt[9][4];
#pragma unroll
    for (int tap = 0; tap < 9; ++tap) {
        const float4 v = *reinterpret_cast<const float4*>(dw_w + (tap * CC + c) * MM);
        wt[tap][0] = v.x; wt[tap][1] = v.y; wt[tap][2] = v.z; wt[tap][3] = v.w;
    }

    float mx0 = -INFINITY, mx1 = -INFINITY, mx2 = -INFINITY, mx3 = -INFINITY;
    for (int t = 0; t < TT; ++t) {
        float a0 = 0.f, a1 = 0.f, a2 = 0.f, a3 = 0.f;
#pragma unroll
        for (int tap = 0; tap < 9; ++tap) {
            const int kh = tap / 3, kw = tap % 3;
            const int hh = h + kh - 1, ww = w + kw - 1;
            if (hh >= 0 && hh < HH && ww >= 0 && ww < WWD) {   // uniform per block
                const float xv = x[(((b * TT + t) * HH + hh) * WWD + ww) * CC + c];
                a0 = fmaf(xv, wt[tap][0], a0);
                a1 = fmaf(xv, wt[tap][1], a1);
                a2 = fmaf(xv, wt[tap][2], a2);
                a3 = fmaf(xv, wt[tap][3], a3);
            }
        }
        mx0 = fmaxf(mx0, a0); mx1 = fmaxf(mx1, a1);
        mx2 = fmaxf(mx2, a2); mx3 = fmaxf(mx3, a3);
    }

    const float4 bias = *reinterpret_cast<const float4*>(dw_b + c * MM);
    float4 o;
    o.x = mx0 + bias.x; o.y = mx1 + bias.y; o.z = mx2 + bias.z; o.w = mx3 + bias.w;
    *reinterpret_cast<float4*>(ymax + ((b * HH + h) * WWD + w) * (CC * MM) + c * MM) = o;
}

// ---------------------------------------------------------------------------
// Kernel 2: per-channel-group 3x3 conv (1024 groups of width 4) via
// V_WMMA_F32_16X16X4_F32.  One wave = (b, c, 16-row spatial tile):
//   D[16 spatial rows, 16 cols(m_out in 0..3)] = sum over 9 taps of A(16x4)*B(4x16)
// A lane layout: VGPR j, half h -> K = 2h+j (m_in), row = lane&15 (spatial).
// B lane layout: VGPR j, half h -> K = 2h+j (m_in), col = lane&15 (m_out).
// D lane layout: VGPR r, half h -> row r+8h, col = lane&15.
// Accumulator pre-seeded with conv bias.
// ---------------------------------------------------------------------------
__global__ __launch_bounds__(256) void gconv_wmma_kernel(
    const float* __restrict__ ymax,    // (B,H,W,C*M) flat dw-channel order
    const float* __restrict__ conv_w,  // (C,3,3,M,M)
    const float* __restrict__ conv_b,  // (C,M)
    float* __restrict__ out)           // (B,M,H,W,C)
{
    const int lane   = threadIdx.x & 31;
    const int waveId = blockIdx.x * (blockDim.x >> 5) + (threadIdx.x >> 5);

    const int c  = waveId & (CC - 1);
    const int b  = (waveId >> 10) & (BB - 1);
    const int rt = waveId >> 14;                 // 0..3 : 16-row spatial tile

    const int half = lane >> 4;
    const int li   = lane & 15;
    const int s    = rt * 16 + li;               // spatial row index for A
    const int sh   = s / WWD, sw = s % WWD;
    const bool rowValid = (s < HH * WWD);

    const int m0 = 2 * half;                     // m_in carried in vgpr 0
    const int m1 = 2 * half + 1;                 // m_in carried in vgpr 1

    // C accumulator = bias (all 16 rows identical -> broadcast into 8 VGPRs)
    const float bias = (li < MM) ? conv_b[c * MM + li] : 0.0f;
    v8f acc;
#pragma unroll
    for (int i = 0; i < 8; ++i) acc[i] = bias;

#pragma unroll
    for (int tap = 0; tap < 9; ++tap) {
        const int kh = tap / 3, kw = tap % 3;

        // A: input activations (mixed channel index m_in*1024 + c)
        v2f a; a[0] = 0.0f; a[1] = 0.0f;
        const int hh = sh + kh - 1, ww = sw + kw - 1;
        if (rowValid && hh >= 0 && hh < HH && ww >= 0 && ww < WWD) {
            const int base = ((b * HH + hh) * WWD + ww) * (CC * MM) + c;
            a[0] = ymax[base + m0 * CC];
            a[1] = ymax[base + m1 * CC];
        }

        // B: weights for this group/tap; columns >= 4 forced to zero
        v2f bm; bm[0] = 0.0f; bm[1] = 0.0f;
        if (li < MM) {
            const int wbase = ((c * 3 + kh) * 3 + kw) * (MM * MM) + li;
            bm[0] = conv_w[wbase + m0 * MM];
            bm[1] = conv_w[wbase + m1 * MM];
        }

        // full wave active here -> EXEC all ones as WMMA requires
        acc = __builtin_amdgcn_wmma_f32_16x16x4_f32(
            /*neg_a=*/false, a, /*neg_b=*/false, bm,
            /*c_mod=*/(short)0, acc, /*reuse_a=*/false, /*reuse_b=*/false);
    }

    // Scatter D: out[b][m_out=li][srow/7][srow%7][c]
    if (li < MM) {
#pragma unroll
        for (int r = 0; r < 8; ++r) {
            const int srow = rt * 16 + r + 8 * half;
            if (srow < HH * WWD) {
                const int oh = srow / WWD, ow = srow % WWD;
                out[(((b * MM + li) * HH + oh) * WWD + ow) * CC + c] = acc[r];
            }
        }
    }
}

// ---------------------------------------------------------------------------
extern "C" void kernel_launch(void* const* d_in, const int* in_sizes, int n_in,
                              void* d_out, int out_size, void* d_ws, size_t ws_size,
                              hipStream_t stream) {
    const float* x      = (const float*)d_in[0];   // (16,20,7,7,1024)
    const float* dw_w   = (const float*)d_in[1];   // (3,3,1024,4)
    const float* dw_b   = (const float*)d_in[2];   // (4096)
    const float* conv_w = (const float*)d_in[3];   // (1024,3,3,4,4)
    const float* conv_b = (const float*)d_in[4];   // (1024,4)
    float* out  = (float*)d_out;                   // (16,4,7,7,1024)
    float* ymax = (float*)d_ws;                    // 16*7*7*4096 f32 = 12.85 MB

    // Stage 1: depthwise conv + temporal max
    dim3 g1(CC / 256, HH * WWD, BB);
    dw_tmax_kernel<<<g1, 256, 0, stream>>>(x, dw_w, dw_b, ymax);

    // Stage 2: grouped conv via WMMA. waves = C * B * 4 row-tiles = 65536
    const int totalWaves = CC * BB * 4;
    gconv_wmma_kernel<<<totalWaves / 8, 256, 0, stream>>>(ymax, conv_w, conv_b, out);
}